// Gate_20486994002329
// MI455X (gfx1250) — compile-verified
//
#include <hip/hip_runtime.h>
#include <math.h>

// ---------------------------------------------------------------------------
// MoE Gate (softmax router, top-2, aux load-balancing loss) for MI455X gfx1250
//
// Bandwidth-bound (134 MB of activations, 0.54 GFLOP -> ~4 FLOP/byte).
// Uses V_WMMA_F32_16X16X4_F32: one wave owns a 16-token x 16-expert tile
// (experts 8..15 masked), K swept in steps of 4 with two accumulators.
// ---------------------------------------------------------------------------

typedef __attribute__((ext_vector_type(2))) float v2f;
typedef __attribute__((ext_vector_type(8))) float v8f;

#define DIM      2048
#define NEXP     8
#define NTOK     16384     // 4 * 4096 tokens
#define TOK_PB   4096      // tokens per batch
#define ALPHA    0.1f
#define OUT_IDX  32768     // idx region offset in d_out (floats)
#define OUT_AUX  65536     // aux scalar offset in d_out

// Zero the 64-float workspace (32 pi-sums + 32 fi-counts) every call:
// harness poisons ws once and never re-poisons between graph replays.
__global__ void gate_zero_ws(float* __restrict__ ws) {
    if (threadIdx.x < 64) ws[threadIdx.x] = 0.0f;
}

// 256 threads = 8 waves; each wave handles 16 tokens; block = 128 tokens
// (always within a single batch since 4096 % 128 == 0).
__global__ __launch_bounds__(256) void gate_main(const float* __restrict__ x,
                                                 const float* __restrict__ W,
                                                 float* __restrict__ out,
                                                 float* __restrict__ ws) {
    __shared__ float sc[8][16][NEXP];   // per-wave 16x8 score tiles
    __shared__ float blkSum[NEXP];      // per-block sum of softmax probs (-> pi)
    __shared__ float blkCnt[NEXP];      // per-block top-2 hit counts   (-> fi)

    const int tid   = threadIdx.x;
    const int wave  = tid >> 5;
    const int lane  = tid & 31;
    const int blockTokBase = blockIdx.x * 128;
    const int tokBase      = blockTokBase + wave * 16;

    // A-matrix (16x4 f32) layout: lane&15 = M row; VGPR0/1 hold K,K+1 where
    // K = (lane>=16 ? 2 : 0).  => one aligned float2 load per lane per WMMA.
    const int m    = lane & 15;             // token row in tile / expert col in B
    const int koff = (lane >> 4) * 2;       // 0 for lanes 0-15, 2 for lanes 16-31

    const float* xrow = x + (size_t)(tokBase + m) * DIM + koff;
    const float* wrow = W + (size_t)(m & 7) * DIM + koff;   // always-valid addr
    const float  bmsk = (m < NEXP) ? 1.0f : 0.0f;           // mask cols 8..15

    v8f c0 = {0.f,0.f,0.f,0.f,0.f,0.f,0.f,0.f};
    v8f c1 = {0.f,0.f,0.f,0.f,0.f,0.f,0.f,0.f};

    for (int k = 0; k < DIM; k += 16) {
        v2f a0 = *(const v2f*)(xrow + k);
        v2f a1 = *(const v2f*)(xrow + k + 4);
        v2f a2 = *(const v2f*)(xrow + k + 8);
        v2f a3 = *(const v2f*)(xrow + k + 12);
        v2f b0 = *(const v2f*)(wrow + k)      * bmsk;
        v2f b1 = *(const v2f*)(wrow + k + 4)  * bmsk;
        v2f b2 = *(const v2f*)(wrow + k + 8)  * bmsk;
        v2f b3 = *(const v2f*)(wrow + k + 12) * bmsk;
        // Two independent accumulator chains to relax WMMA->WMMA latency.
        c0 = __builtin_amdgcn_wmma_f32_16x16x4_f32(false, a0, false, b0,
                                                   (short)0, c0, false, false);
        c1 = __builtin_amdgcn_wmma_f32_16x16x4_f32(false, a1, false, b1,
                                                   (short)0, c1, false, false);
        c0 = __builtin_amdgcn_wmma_f32_16x16x4_f32(false, a2, false, b2,
                                                   (short)0, c0, false, false);
        c1 = __builtin_amdgcn_wmma_f32_16x16x4_f32(false, a3, false, b3,
                                                   (short)0, c1, false, false);
    }
    v8f c = c0 + c1;

    // C/D layout: VGPR v -> (M=v, N=lane) for lanes 0-15, (M=v+8, N=lane-16)
    // for lanes 16-31.  Keep only the 8 real expert columns.
    #pragma unroll
    for (int v = 0; v < 8; ++v) {
        const int M = (lane < 16) ? v : (v + 8);
        if (m < NEXP) sc[wave][M][m] = c[v];
    }
    if (tid < NEXP) { blkSum[tid] = 0.0f; blkCnt[tid] = 0.0f; }
    __syncthreads();

    if (lane < 16) {
        float s[NEXP];
        float mx = -INFINITY;
        #pragma unroll
        for (int e = 0; e < NEXP; ++e) { s[e] = sc[wave][lane][e]; mx = fmaxf(mx, s[e]); }
        float sum = 0.0f;
        #pragma unroll
        for (int e = 0; e < NEXP; ++e) { s[e] = __expf(s[e] - mx); sum += s[e]; }
        const float inv = 1.0f / sum;
        #pragma unroll
        for (int e = 0; e < NEXP; ++e) s[e] *= inv;

        // top-2 with lowest-index tie-break (matches jax.lax.top_k)
        int i0 = 0; float p0 = s[0];
        #pragma unroll
        for (int e = 1; e < NEXP; ++e) if (s[e] > p0) { p0 = s[e]; i0 = e; }
        int i1 = -1; float p1 = -INFINITY;
        #pragma unroll
        for (int e = 0; e < NEXP; ++e) if (e != i0 && s[e] > p1) { p1 = s[e]; i1 = e; }

        const int tok = tokBase + lane;
        out[(size_t)tok * 2 + 0]           = p0;          // ROUTE_SCALE == 1.0
        out[(size_t)tok * 2 + 1]           = p1;
        out[OUT_IDX + (size_t)tok * 2 + 0] = (float)i0;
        out[OUT_IDX + (size_t)tok * 2 + 1] = (float)i1;

        #pragma unroll
        for (int e = 0; e < NEXP; ++e) atomicAdd(&blkSum[e], s[e]);
        atomicAdd(&blkCnt[i0], 1.0f);
        atomicAdd(&blkCnt[i1], 1.0f);
    }
    __syncthreads();

    if (tid < NEXP) {
        const int batch = blockTokBase / TOK_PB;
        atomicAdd(&ws[batch * NEXP + tid],      blkSum[tid]);  // pi accumulator
        atomicAdd(&ws[32 + batch * NEXP + tid], blkCnt[tid]);  // fi accumulator
    }
}

__global__ void gate_aux(const float* __restrict__ ws, float* __restrict__ out) {
    if (threadIdx.x == 0) {
        float aux = 0.0f;
        for (int b = 0; b < 4; ++b) {
            float acc = 0.0f;
            for (int e = 0; e < NEXP; ++e) {
                const float pi = ws[b * NEXP + e] * (1.0f / (float)TOK_PB);
                const float fi = ws[32 + b * NEXP + e] *
                                 ((float)NEXP / (2.0f * (float)TOK_PB));
                acc += fi * pi;
            }
            aux += acc;
        }
        out[OUT_AUX] = aux * 0.25f * ALPHA;   // mean over 4 batches, * alpha
    }
}

extern "C" void kernel_launch(void* const* d_in, const int* in_sizes, int n_in,
                              void* d_out, int out_size, void* d_ws, size_t ws_size,
                              hipStream_t stream) {
    const float* x = (const float*)d_in[0];   // (4, 4096, 2048) f32
    const float* W = (const float*)d_in[1];   // (8, 2048) f32
    float* out = (float*)d_out;               // [weight|idx|aux] = 65537 f32
    float* ws  = (float*)d_ws;                // 64 f32 accumulators

    gate_zero_ws<<<1, 64, 0, stream>>>(ws);
    gate_main<<<NTOK / 128, 256, 0, stream>>>(x, W, out, ws);
    gate_aux<<<1, 32, 0, stream>>>(ws, out);
}